// _GCNLayer_59279138619813
// MI455X (gfx1250) — compile-verified
//
#include <hip/hip_runtime.h>
#include <math.h>

#define D 64
#define LN_EPS 1e-5f
#define DEG_EPS 1e-12f

typedef __attribute__((ext_vector_type(2))) float v2f;
typedef __attribute__((ext_vector_type(8))) float v8f;

// ---------------- Kernel 1: init degree with self-loop weight ----------------
__global__ void k_init_deg(float* __restrict__ deg, int N) {
    int i = blockIdx.x * blockDim.x + threadIdx.x;
    if (i < N) deg[i] = 1.0f;   // self-loop contributes weight 1.0
}

// ---------------- Kernel 2: accumulate edge weights into degree --------------
__global__ void k_deg_accum(const long long* __restrict__ row,
                            const float* __restrict__ ew,
                            float* __restrict__ deg, int E) {
    int e = blockIdx.x * blockDim.x + threadIdx.x;
    if (e < E) atomicAdd(&deg[(int)row[e]], ew[e]);
}

// ------- Kernel 3: dis = rsqrt(deg+eps); h = x * dis^2 (self-loop term) ------
__global__ void k_dis_selfloop(const float* __restrict__ deg,
                               const float* __restrict__ x,
                               float* __restrict__ dis,
                               float* __restrict__ h, int N) {
    int gid = blockIdx.x * blockDim.x + threadIdx.x;
    int i = gid >> 4;        // node
    int q = gid & 15;        // which float4 of the 64-float row
    if (i >= N) return;
    float dv = rsqrtf(deg[i] + DEG_EPS);
    if (q == 0) dis[i] = dv;
    float s = dv * dv;       // normalized self-loop weight
    float4 v = ((const float4*)(x + (size_t)i * D))[q];
    float4 r; r.x = v.x * s; r.y = v.y * s; r.z = v.z * s; r.w = v.w * s;
    ((float4*)(h + (size_t)i * D))[q] = r;
}

// ---------------- Kernel 4: edge SpMM via f32 global atomics -----------------
__global__ void k_spmm(const long long* __restrict__ row,
                       const long long* __restrict__ col,
                       const float* __restrict__ ew,
                       const float* __restrict__ dis,
                       const float* __restrict__ x,
                       float* __restrict__ h, int E) {
    long long gid = (long long)blockIdx.x * blockDim.x + threadIdx.x;
    int e = (int)(gid >> 4);
    int q = (int)(gid & 15);
    if (e >= E) return;
    int r = (int)row[e];
    int c = (int)col[e];
    float nw = ew[e] * dis[r] * dis[c];
    float4 v = ((const float4*)(x + (size_t)c * D))[q];
    float* hp = h + (size_t)r * D + q * 4;
    atomicAdd(hp + 0, v.x * nw);
    atomicAdd(hp + 1, v.y * nw);
    atomicAdd(hp + 2, v.z * nw);
    atomicAdd(hp + 3, v.w * nw);
}

// ------- Kernel 5: fused  y = LN(gelu(h@W^T + b))*g + beta + x  (WMMA) -------
// One wave32 per 16-row tile. y tile is 16x64 = four 16x16 column tiles,
// each accumulated over K=64 with 16x V_WMMA_F32_16X16X4_F32.
//
// Fragment layouts (ISA 7.12.2, 32-bit):
//   A (16x4 MxK):  lane16 = M row; v0 = K {0 | 2 for upper half}, v1 = K {1|3}
//       -> contiguous float2 load from h row at col kbase + 2*half.
//   B (4x16 KxN):  lane16 = N col; v0 = K {0|2}, v1 = K {1|3}
//       -> B[k][j] = W[j][k], so contiguous float2 load from W row j.
//   C/D (16x16):   lane16 = N col; VGPR r holds row M = r + 8*half.
__global__ __launch_bounds__(256) void k_wmma_fused(
        const float* __restrict__ h, const float* __restrict__ x,
        const float* __restrict__ Wt, const float* __restrict__ bias,
        const float* __restrict__ gamma, const float* __restrict__ beta,
        float* __restrict__ out, int ntiles) {
    int wave = (int)((blockIdx.x * blockDim.x + threadIdx.x) >> 5);
    int lane = threadIdx.x & 31;
    if (wave >= ntiles) return;             // wave-uniform: EXEC stays all-ones
    int row0 = wave * 16;
    int half = lane >> 4;
    int l16  = lane & 15;

    // Preload all 16 A fragments (reused across the 4 column tiles).
    v2f a[16];
    const float* hrow = h + (size_t)(row0 + l16) * D + 2 * half;
#pragma unroll
    for (int ks = 0; ks < 16; ++ks)
        a[ks] = *(const v2f*)(hrow + 4 * ks);

    v8f acc[4];
    float gj[4], bj[4], biasj[4];
#pragma unroll
    for (int jt = 0; jt < 4; ++jt) {
        int j = jt * 16 + l16;
        biasj[jt] = bias[j];
        gj[jt]    = gamma[j];
        bj[jt]    = beta[j];
        const float* wrow = Wt + (size_t)j * D + 2 * half;  // row j of W == col j of W^T
        v8f c = {};
#pragma unroll
        for (int ks = 0; ks < 16; ++ks) {
            v2f b = *(const v2f*)(wrow + 4 * ks);
            c = __builtin_amdgcn_wmma_f32_16x16x4_f32(
                    /*neg_a=*/false, a[ks], /*neg_b=*/false, b,
                    /*c_mod=*/(short)0, c, /*reuse_a=*/false, /*reuse_b=*/false);
        }
        acc[jt] = c;
    }

    // bias + exact-erf GELU, and per-row partial sums for LayerNorm
    float s1[8], s2[8];
#pragma unroll
    for (int r = 0; r < 8; ++r) { s1[r] = 0.0f; s2[r] = 0.0f; }
#pragma unroll
    for (int jt = 0; jt < 4; ++jt) {
#pragma unroll
        for (int r = 0; r < 8; ++r) {
            float v = acc[jt][r] + biasj[jt];
            v = 0.5f * v * (1.0f + erff(v * 0.70710678118654752f));
            acc[jt][r] = v;
            s1[r] += v;
            s2[r] += v * v;
        }
    }
    // Butterfly reduction over the 16 lanes of each wave32 half
    // (masks 1/2/4/8 never flip bit 4, so halves stay independent).
#pragma unroll
    for (int m = 1; m < 16; m <<= 1) {
#pragma unroll
        for (int r = 0; r < 8; ++r) {
            s1[r] += __shfl_xor(s1[r], m, 32);
            s2[r] += __shfl_xor(s2[r], m, 32);
        }
    }

    float mean[8], rstd[8];
#pragma unroll
    for (int r = 0; r < 8; ++r) {
        float mu = s1[r] * (1.0f / 64.0f);
        float var = s2[r] * (1.0f / 64.0f) - mu * mu;
        mean[r] = mu;
        rstd[r] = rsqrtf(var + LN_EPS);
    }

    // normalize, affine, residual, store
#pragma unroll
    for (int jt = 0; jt < 4; ++jt) {
        int j = jt * 16 + l16;
#pragma unroll
        for (int r = 0; r < 8; ++r) {
            int rowm = row0 + r + 8 * half;
            size_t idx = (size_t)rowm * D + j;
            float v = (acc[jt][r] - mean[r]) * rstd[r] * gj[jt] + bj[jt] + x[idx];
            out[idx] = v;
        }
    }
}

extern "C" void kernel_launch(void* const* d_in, const int* in_sizes, int n_in,
                              void* d_out, int out_size, void* d_ws, size_t ws_size,
                              hipStream_t stream) {
    const float*     x     = (const float*)d_in[0];
    const long long* ei    = (const long long*)d_in[1];   // int64 (2, E)
    const float*     ew    = (const float*)d_in[2];
    const float*     W     = (const float*)d_in[3];
    const float*     bias  = (const float*)d_in[4];
    const float*     gamma = (const float*)d_in[5];
    const float*     beta  = (const float*)d_in[6];
    float*           out   = (float*)d_out;

    int N = in_sizes[0] / D;
    int E = in_sizes[2];
    const long long* row = ei;
    const long long* col = ei + E;

    // workspace layout: deg[N] | dis[N] | h[N*64]
    float* deg = (float*)d_ws;
    float* dis = deg + N;
    float* h   = dis + N;

    k_init_deg<<<(N + 255) / 256, 256, 0, stream>>>(deg, N);
    k_deg_accum<<<(E + 255) / 256, 256, 0, stream>>>(row, ew, deg, E);

    long long t3 = (long long)N * 16;                 // 16 threads per node row
    k_dis_selfloop<<<(unsigned)((t3 + 255) / 256), 256, 0, stream>>>(deg, x, dis, h, N);

    long long t4 = (long long)E * 16;                 // 16 threads per edge
    k_spmm<<<(unsigned)((t4 + 255) / 256), 256, 0, stream>>>(row, col, ew, dis, x, h, E);

    int ntiles = (N + 15) / 16;                       // N=100000 -> exactly 6250 tiles
    int blocks = (ntiles + 7) / 8;                    // 8 waves (one tile each) per block
    k_wmma_fused<<<blocks, 256, 0, stream>>>(h, x, W, bias, gamma, beta, out, ntiles);
}